// GIoULoss_44538810860250
// MI455X (gfx1250) — compile-verified
//
#include <hip/hip_runtime.h>
#include <hip/hip_bf16.h>

#define EPS 1e-6f

typedef __attribute__((ext_vector_type(2))) float v2f;
typedef __attribute__((ext_vector_type(8))) float v8f;

// Wave32 sum-reduction via V_WMMA_F32_16X16X4_F32.
// A: lane's partial in slot0, 0 in slot1 -> sum(all A entries) == sum over lanes.
// B: all ones -> D[m][n] = sum_k A[m][k]; summing one column of D
// (8 regs of lane n plus 8 regs of lane n+16) gives the total, independent of
// the exact A-slot permutation. EXEC must be all-1s (callers are convergent).
static __device__ __forceinline__ float wave_reduce_wmma(float s) {
    v2f a; a[0] = s;    a[1] = 0.0f;
    v2f b; b[0] = 1.0f; b[1] = 1.0f;
    v8f c = {};
    c = __builtin_amdgcn_wmma_f32_16x16x4_f32(
        /*neg_a=*/false, a, /*neg_b=*/false, b,
        /*c_mod=*/(short)0, c, /*reuse_a=*/false, /*reuse_b=*/false);
    float r = c[0] + c[1] + c[2] + c[3] + c[4] + c[5] + c[6] + c[7];
    r += __shfl_xor(r, 16, 32);   // pair lane n with lane n+16 -> full column sum
    return r;                      // lanes 0..15 hold the full 32-lane total
}

__global__ __launch_bounds__(256)
void giou_partial_kernel(const float4* __restrict__ pred,
                         const float4* __restrict__ targ,
                         float* __restrict__ block_sums,
                         int n, float scale) {
    __shared__ float wsum[8];
    const int tid = threadIdx.x;
    const int stride = (int)gridDim.x * 256;      // 32-bit: n <= 4M
    float s = 0.0f;

    for (int i = (int)blockIdx.x * 256 + tid; i < n; i += stride) {
        // Unconditional speculative prefetch (TH=0): dropped silently if the
        // address doesn't translate, so no bounds guard / branch needed.
        __builtin_prefetch(&pred[i + stride], 0, 0);
        __builtin_prefetch(&targ[i + stride], 0, 0);

        const float4 p = pred[i];
        const float4 t = targ[i];

        // cxcywh -> xyxy
        const float px1 = p.x - 0.5f * p.z, py1 = p.y - 0.5f * p.w;
        const float px2 = p.x + 0.5f * p.z, py2 = p.y + 0.5f * p.w;
        const float tx1 = t.x - 0.5f * t.z, ty1 = t.y - 0.5f * t.w;
        const float tx2 = t.x + 0.5f * t.z, ty2 = t.y + 0.5f * t.w;

        const float area_p = (px2 - px1) * (py2 - py1);
        const float area_t = (tx2 - tx1) * (ty2 - ty1);

        const float iw = fmaxf(fminf(px2, tx2) - fmaxf(px1, tx1), 0.0f);
        const float ih = fmaxf(fminf(py2, ty2) - fmaxf(py1, ty1), 0.0f);
        const float inter = iw * ih;
        const float uni   = area_p + area_t - inter;
        const float iou   = inter / fmaxf(uni, EPS);

        const float cw = fmaxf(fmaxf(px2, tx2) - fminf(px1, tx1), 0.0f);
        const float ch = fmaxf(fmaxf(py2, ty2) - fminf(py1, ty1), 0.0f);
        const float area_c = cw * ch;

        const float giou = iou - (area_c - uni) / fmaxf(area_c, EPS);
        s += 1.0f - giou;
    }

    // convergent: all 256 threads reach here with EXEC all-1s
    const float r = wave_reduce_wmma(s);
    const int lane = tid & 31, wave = tid >> 5;
    if (lane == 0) wsum[wave] = r;
    __syncthreads();
    if (tid == 0) {
        float b = 0.0f;
        #pragma unroll
        for (int w = 0; w < 8; ++w) b += wsum[w];
        block_sums[blockIdx.x] = b * scale;
    }
}

__global__ __launch_bounds__(256)
void giou_final_kernel(const float* __restrict__ partials, int count,
                       float* __restrict__ out, float inv_n) {
    __shared__ float wsum[8];
    float s = 0.0f;
    for (int i = threadIdx.x; i < count; i += 256) s += partials[i];

    const float r = wave_reduce_wmma(s);
    const int lane = threadIdx.x & 31, wave = threadIdx.x >> 5;
    if (lane == 0) wsum[wave] = r;
    __syncthreads();
    if (threadIdx.x == 0) {
        float b = 0.0f;
        #pragma unroll
        for (int w = 0; w < 8; ++w) b += wsum[w];
        out[0] = b * inv_n;
    }
}

extern "C" void kernel_launch(void* const* d_in, const int* in_sizes, int n_in,
                              void* d_out, int out_size, void* d_ws, size_t ws_size,
                              hipStream_t stream) {
    const float4* pred = (const float4*)d_in[0];
    const float4* targ = (const float4*)d_in[1];
    float* out = (float*)d_out;

    const int n = in_sizes[0] / 4;                 // boxes (flat count / 4)
    const float inv_n = (float)(1.0 / (double)n);

    const int max_ws_blocks = (int)(ws_size / sizeof(float));
    if (max_ws_blocks >= 64) {
        int nblocks = max_ws_blocks < 2048 ? max_ws_blocks : 2048;
        giou_partial_kernel<<<nblocks, 256, 0, stream>>>(
            pred, targ, (float*)d_ws, n, 1.0f);
        giou_final_kernel<<<1, 256, 0, stream>>>(
            (const float*)d_ws, nblocks, out, inv_n);
    } else {
        // tiny-workspace fallback: single block writes the scaled mean directly
        giou_partial_kernel<<<1, 256, 0, stream>>>(pred, targ, out, n, inv_n);
    }
}